// ASPCaps_62457414418697
// MI455X (gfx1250) — compile-verified
//
#include <hip/hip_runtime.h>
#include <hip/hip_bf16.h>
#include <math.h>

// Problem constants (from reference)
//  B=16, H=64, W=64, C=CI*NI=32, KS=9, K=288, N_OUT=128 (CJ=32,NJ=4)
#define NB   16
#define NH   64
#define NW   64
#define NC   32
#define NKS  9
#define KDIM 288       // 9*32
#define NOUT 128
#define NPIX (NB*NH*NW)   // 65536
#define EPSQ 1e-7f

typedef float v2f __attribute__((ext_vector_type(2)));
typedef float v8f __attribute__((ext_vector_type(8)));

// ---------------------------------------------------------------------------
// Kernel 1: 3x3 SAME offset conv: x[B,H,W,32] * Wof[3,3,32,27] + bof -> offs[P,27]
// ---------------------------------------------------------------------------
__global__ __launch_bounds__(256) void aspcaps_offset_conv(
    const float* __restrict__ x, const float* __restrict__ Wof,
    const float* __restrict__ bof, float* __restrict__ offs)
{
    int idx = blockIdx.x * 256 + threadIdx.x;
    if (idx >= NPIX * 27) return;
    int p  = idx / 27;
    int oc = idx - p * 27;
    int b  = p >> 12;
    int hw = p & 4095;
    int h  = hw >> 6;
    int w  = hw & 63;

    float acc = bof[oc];
    #pragma unroll
    for (int ky = 0; ky < 3; ++ky) {
        int y = h + ky - 1;
        if ((unsigned)y >= (unsigned)NH) continue;
        #pragma unroll
        for (int kx = 0; kx < 3; ++kx) {
            int xx = w + kx - 1;
            if ((unsigned)xx >= (unsigned)NW) continue;
            const float* xp = x + (((size_t)(b * NH + y)) * NW + xx) * NC;
            const float* wp = Wof + ((size_t)((ky * 3 + kx) * NC)) * 27 + oc;
            #pragma unroll 8
            for (int c = 0; c < NC; ++c)
                acc = fmaf(xp[c], wp[(size_t)c * 27], acc);
        }
    }
    offs[(size_t)p * 27 + oc] = acc;
}

// ---------------------------------------------------------------------------
// Kernel 2: deformable bilinear sampling + sigmoid modulation -> map_all[P,288]
// One thread per (pixel, sampling point k); 32 channels via float4 loads.
// ---------------------------------------------------------------------------
__global__ __launch_bounds__(256) void aspcaps_sample(
    const float* __restrict__ x, const float* __restrict__ offs,
    float* __restrict__ map_all)
{
    int idx = blockIdx.x * 256 + threadIdx.x;
    if (idx >= NPIX * NKS) return;
    int p = idx / NKS;
    int k = idx - p * NKS;
    int b  = p >> 12;
    int hw = p & 4095;
    int h  = hw >> 6;
    int w  = hw & 63;

    const float* o = offs + (size_t)p * 27;
    float offy = o[2 * k];
    float offx = o[2 * k + 1];
    float mval = 1.0f / (1.0f + expf(-o[18 + k]));

    int py = k / 3 - 1;
    int px = k - (k / 3) * 3 - 1;
    // padded-coordinate grid (pad=1), as in reference
    float gy = (float)(h + 1 + py) + offy;
    float gx = (float)(w + 1 + px) + offx;
    const float hiy = (float)(NH + 1);   // 65
    const float hix = (float)(NW + 1);   // 65
    float fy = floorf(gy), fx = floorf(gx);
    float y0 = fminf(fmaxf(fy,        0.f), hiy);
    float y1 = fminf(fmaxf(fy + 1.f,  0.f), hiy);
    float x0 = fminf(fmaxf(fx,        0.f), hix);
    float x1 = fminf(fmaxf(fx + 1.f,  0.f), hix);
    float gyc = fminf(fmaxf(gy, 0.f), hiy);
    float gxc = fminf(fmaxf(gx, 0.f), hix);
    float dy0 = gyc - y0, dx0 = gxc - x0;
    float dy1 = y1 - gyc, dx1 = x1 - gxc;
    // corner order matches reference: (y1,x1),(y1,x0),(y0,x1),(y0,x0)
    float w11 = dy0 * dx0, w10 = dy0 * dx1, w01 = dy1 * dx0, w00 = dy1 * dx1;
    int iy0 = (int)y0, iy1 = (int)y1, ix0 = (int)x0, ix1 = (int)x1;

    auto cptr = [&](int yy, int xx) -> const float* {
        if (yy < 1 || yy > NH || xx < 1 || xx > NW) return (const float*)0;
        return x + (((size_t)(b * NH + (yy - 1))) * NW + (xx - 1)) * NC;
    };
    const float* p11 = cptr(iy1, ix1);
    const float* p10 = cptr(iy1, ix0);
    const float* p01 = cptr(iy0, ix1);
    const float* p00 = cptr(iy0, ix0);

    float* dst = map_all + (size_t)p * KDIM + k * NC;
    const float4 z4 = {0.f, 0.f, 0.f, 0.f};
    #pragma unroll
    for (int c = 0; c < NC; c += 4) {
        float4 a11 = p11 ? *(const float4*)(p11 + c) : z4;
        float4 a10 = p10 ? *(const float4*)(p10 + c) : z4;
        float4 a01 = p01 ? *(const float4*)(p01 + c) : z4;
        float4 a00 = p00 ? *(const float4*)(p00 + c) : z4;
        float4 r;
        r.x = mval * (w11 * a11.x + w10 * a10.x + w01 * a01.x + w00 * a00.x);
        r.y = mval * (w11 * a11.y + w10 * a10.y + w01 * a01.y + w00 * a00.y);
        r.z = mval * (w11 * a11.z + w10 * a10.z + w01 * a01.z + w00 * a00.z);
        r.w = mval * (w11 * a11.w + w10 * a10.w + w01 * a01.w + w00 * a00.w);
        *(float4*)(dst + c) = r;
    }
}

// ---------------------------------------------------------------------------
// Kernel 3: WMMA GEMM [32 x 288] @ [288 x 128] + bias, then capsule squash.
// Block = 256 threads = 8 wave32; wave w owns output columns [16w, 16w+16)
// and computes TWO 16x16 M-subtiles per k-step, reusing one B fragment.
// A tile (32x288) staged in LDS, row stride 290 floats (8B-aligned rows).
// v_wmma_f32_16x16x4_f32, 72 k-steps x 2 subtiles = 144 WMMAs/wave.
// EXEC all-1s throughout (no divergence), as the ISA requires.
// ---------------------------------------------------------------------------
#define A_STRIDE 290
#define MTILE    32

__global__ __launch_bounds__(256) void aspcaps_gemm_squash(
    const float* __restrict__ map_all, const float* __restrict__ Wmat,
    const float* __restrict__ bias, float* __restrict__ out)
{
    __shared__ float Alds[MTILE * A_STRIDE];   // 37,120 B
    const int tile = blockIdx.x;               // 2048 tiles of 32 pixels
    const int t    = threadIdx.x;

    // cooperative load of the 32x288 A tile
    const float* Ag = map_all + (size_t)tile * MTILE * KDIM;
    for (int i = t; i < MTILE * KDIM; i += 256) {
        int m = i / KDIM;
        int k = i - m * KDIM;
        Alds[m * A_STRIDE + k] = Ag[i];
    }
    __syncthreads();

    const int wave = t >> 5;
    const int lane = t & 31;
    const int half = lane >> 4;   // 0: K lo pair, 1: K hi pair
    const int l15  = lane & 15;
    const int n0   = wave * 16;

    // Bias folded into accumulator init: C/D element (r,lane) sits at column
    // n = n0 + (lane&15) for every VGPR row r, so a lane-constant init is exact.
    const float bv = bias[n0 + l15];
    v8f acc0 = {bv, bv, bv, bv, bv, bv, bv, bv};   // pixels tile*32 + [0,16)
    v8f acc1 = {bv, bv, bv, bv, bv, bv, bv, bv};   // pixels tile*32 + [16,32)

    const float* arow0 = Alds + l15 * A_STRIDE;
    const float* arow1 = Alds + (l15 + 16) * A_STRIDE;
    #pragma unroll 4
    for (int kk = 0; kk < KDIM / 4; ++kk) {
        const int ka = kk * 4 + half * 2;
        // A fragments: 16x4 f32; lanes 0-15 hold K={k,k+1}, lanes 16-31 K={k+2,k+3}
        v2f a0 = *(const v2f*)(arow0 + ka);        // 8B-aligned (even stride, even ka)
        v2f a1 = *(const v2f*)(arow1 + ka);
        // B fragment: 4x16 f32, row-striped across lanes; shared by both subtiles
        const float* bp = Wmat + (size_t)ka * NOUT + n0 + l15;
        v2f b;
        b.x = bp[0];
        b.y = bp[NOUT];
        acc0 = __builtin_amdgcn_wmma_f32_16x16x4_f32(
                   false, a0, false, b, (short)0, acc0, false, false);
        acc1 = __builtin_amdgcn_wmma_f32_16x16x4_f32(
                   false, a1, false, b, (short)0, acc1, false, false);
    }

    // Epilogue: capsule squash over groups of 4 consecutive output channels.
    // C/D layout: VGPR r, lane L -> m = r + (L>=16 ? 8 : 0), n = L&15.
    // A channel group of 4 == 4 consecutive lanes with identical m, so two
    // shfl_xor steps reduce the squared norm within each capsule.
    const int pbase0 = tile * MTILE + half * 8;
    const int pbase1 = pbase0 + 16;
    #pragma unroll
    for (int r = 0; r < 8; ++r) {
        float v  = acc0[r];
        float sq = v * v;
        sq += __shfl_xor(sq, 1, 32);
        sq += __shfl_xor(sq, 2, 32);
        float s = sq / ((1.0f + sq) * sqrtf(sq + EPSQ));
        out[(size_t)(pbase0 + r) * NOUT + n0 + l15] = v * s;

        float v1  = acc1[r];
        float sq1 = v1 * v1;
        sq1 += __shfl_xor(sq1, 1, 32);
        sq1 += __shfl_xor(sq1, 2, 32);
        float s1 = sq1 / ((1.0f + sq1) * sqrtf(sq1 + EPSQ));
        out[(size_t)(pbase1 + r) * NOUT + n0 + l15] = v1 * s1;
    }
}

// ---------------------------------------------------------------------------
extern "C" void kernel_launch(void* const* d_in, const int* in_sizes, int n_in,
                              void* d_out, int out_size, void* d_ws, size_t ws_size,
                              hipStream_t stream) {
    const float* x     = (const float*)d_in[0];   // [16,64,64,8,4] == [P,32]
    const float* kern  = (const float*)d_in[1];   // [3,3,32,128] == [288,128]
    const float* bias  = (const float*)d_in[2];   // [128]
    const float* okern = (const float*)d_in[3];   // [3,3,32,27]
    const float* obias = (const float*)d_in[4];   // [27]
    float* out = (float*)d_out;                   // [P,128]

    // workspace layout: offsets [P,27] then map_all [P,288]
    float* offs    = (float*)d_ws;
    float* map_all = (float*)((char*)d_ws + (size_t)NPIX * 27 * sizeof(float));

    const int n1 = NPIX * 27;
    aspcaps_offset_conv<<<(n1 + 255) / 256, 256, 0, stream>>>(x, okern, obias, offs);

    const int n2 = NPIX * NKS;
    aspcaps_sample<<<(n2 + 255) / 256, 256, 0, stream>>>(x, offs, map_all);

    aspcaps_gemm_squash<<<NPIX / MTILE, 256, 0, stream>>>(map_all, kern, bias, out);
}